// CayleyChaosLoss_46523085750330
// MI455X (gfx1250) — compile-verified
//
#include <hip/hip_runtime.h>
#include <hip/hip_bf16.h>

typedef __attribute__((ext_vector_type(2))) float v2f;
typedef __attribute__((ext_vector_type(8))) float v8f;

#define WAVES_PER_BLOCK 8
#define BATCHES_PER_WAVE 3
#define BATCHES_PER_BLOCK (WAVES_PER_BLOCK * BATCHES_PER_WAVE)

// Fully-unrolled LU determinant with compare-swap partial pivoting.
// All indices are compile-time constants after unrolling -> stays in VGPRs.
template <int N>
__device__ __forceinline__ float det_lu(float (&M)[N][N]) {
    float det = 1.0f;
#pragma unroll
    for (int c = 0; c < N; ++c) {
#pragma unroll
        for (int r = c + 1; r < N; ++r) {
            if (fabsf(M[r][c]) > fabsf(M[c][c])) {
#pragma unroll
                for (int k = c; k < N; ++k) {
                    float t = M[c][k]; M[c][k] = M[r][k]; M[r][k] = t;
                }
                det = -det;
            }
        }
        float piv = M[c][c];
        det *= piv;
        if (piv != 0.0f) {
            float inv = 1.0f / piv;
#pragma unroll
            for (int r = c + 1; r < N; ++r) {
                float f = M[r][c] * inv;
#pragma unroll
                for (int k = c + 1; k < N; ++k)
                    M[r][k] = fmaf(-f, M[c][k], M[r][k]);
            }
        }
    }
    return det;
}

__global__ __launch_bounds__(256) void cayley_main_kernel(
    const float* __restrict__ X, float* __restrict__ partial,
    int B, long long totalRows) {
    __shared__ float sGram[WAVES_PER_BLOCK][16][16];
    __shared__ float sC[BATCHES_PER_BLOCK];
    __shared__ float sE[BATCHES_PER_BLOCK];
    __shared__ float sGm[BATCHES_PER_BLOCK];

    const int tid  = threadIdx.x;
    const int lane = tid & 31;
    const int wv   = tid >> 5;
    const long long gwave = (long long)blockIdx.x * WAVES_PER_BLOCK + wv;

    // WMMA f32 16x16x4 A-operand layout: lanes 0-15 hold row=lane, K={k,k+1};
    // lanes 16-31 hold row=lane-16, K={k+2,k+3}. For G = A*A^T the B-operand
    // layout of A^T is identical, so the same registers feed both A and B.
    const int row  = lane & 15;
    const int koff = (lane >> 4) << 1;
    long long grow = gwave * 15 + row;          // 3 batches = 15 contiguous rows
    if (grow > totalRows - 1) grow = totalRows - 1;  // clamp; garbage rows unused
    const float* p = X + grow * 256 + koff;

    v8f c0 = {0.f, 0.f, 0.f, 0.f, 0.f, 0.f, 0.f, 0.f};
    v8f c1 = c0;
#pragma unroll 4
    for (int k = 0; k < 256; k += 8) {
        v2f a0 = *(const v2f*)(p + k);
        v2f a1 = *(const v2f*)(p + k + 4);
        c0 = __builtin_amdgcn_wmma_f32_16x16x4_f32(false, a0, false, a0,
                                                   (short)0, c0, false, false);
        c1 = __builtin_amdgcn_wmma_f32_16x16x4_f32(false, a1, false, a1,
                                                   (short)0, c1, false, false);
    }
    v8f c = c0 + c1;

    // C/D layout: VGPR g holds row (g or g+8), col = lane%16.
    const int rbase = (lane >> 4) << 3;
#pragma unroll
    for (int g = 0; g < 8; ++g)
        sGram[wv][rbase + g][lane & 15] = c[g];

    __syncthreads();

    // One thread per batch computes the three loss terms from its 5x5 Gram.
    if (tid < BATCHES_PER_BLOCK) {
        const int w2  = tid / 3;
        const int sub = tid - 3 * w2;
        const long long batch = (long long)blockIdx.x * BATCHES_PER_BLOCK + tid;
        float cterm = 0.f, eterm = 0.f, gterm = 0.f;
        if (batch < (long long)B) {
            const int base = 5 * sub;
            float G5[5][5];
#pragma unroll
            for (int i = 0; i < 5; ++i)
#pragma unroll
                for (int j = 0; j < 5; ++j)
                    G5[i][j] = sGram[w2][base + i][base + j];

            // Cayley-Menger matrix: distsq[i][j] = Gii + Gjj - 2Gij
            float M6[6][6];
            M6[0][0] = 0.f;
#pragma unroll
            for (int i = 0; i < 5; ++i) { M6[0][i + 1] = 1.f; M6[i + 1][0] = 1.f; }
#pragma unroll
            for (int i = 0; i < 5; ++i)
#pragma unroll
                for (int j = 0; j < 5; ++j)
                    M6[i + 1][j + 1] = G5[i][i] + G5[j][j] - 2.f * G5[i][j];

            // Edge stats over triu(5,1) order (row-major i<j), ddof=1
            float e[10];
            float esum = 0.f;
            int idx = 0;
#pragma unroll
            for (int i = 0; i < 5; ++i)
#pragma unroll
                for (int j = i + 1; j < 5; ++j) {
                    e[idx] = M6[i + 1][j + 1];
                    esum += e[idx];
                    ++idx;
                }
            float emean = esum * 0.1f;
            float var = 0.f;
#pragma unroll
            for (int q = 0; q < 10; ++q) { float d = e[q] - emean; var += d * d; }
            float estd = sqrtf(var * (1.0f / 9.0f));
            eterm = estd / fmaxf(emean, 1e-6f);

            // Centered Gram: C[ij] = G[ij] - (ri+rj)/5 + s/25
            float r[5]; float s = 0.f;
#pragma unroll
            for (int i = 0; i < 5; ++i) {
                float ri = 0.f;
#pragma unroll
                for (int j = 0; j < 5; ++j) ri += G5[i][j];
                r[i] = ri; s += ri;
            }
            float C5[5][5];
#pragma unroll
            for (int i = 0; i < 5; ++i)
#pragma unroll
                for (int j = 0; j < 5; ++j)
                    C5[i][j] = G5[i][j] - 0.2f * (r[i] + r[j]) + 0.04f * s;
            float trace = C5[0][0] + C5[1][1] + C5[2][2] + C5[3][3] + C5[4][4];
            float det5 = det_lu<5>(C5);
            float cond = det5 / fmaxf(trace, 1e-6f);
            gterm = fmaxf(1.f - cond, 0.f);

            float det6 = det_lu<6>(M6);
            float vsq = fmaxf(-det6 * (1.0f / 9216.0f), 0.f);
            float vol = sqrtf(vsq);
            cterm = fmaxf(1e-4f - vol, 0.f);
        }
        sC[tid] = cterm; sE[tid] = eterm; sGm[tid] = gterm;
    }
    __syncthreads();

    // Fixed-order per-block reduction -> deterministic
    if (tid < 3) {
        const float* src = (tid == 0) ? sC : (tid == 1) ? sE : sGm;
        float acc = 0.f;
#pragma unroll
        for (int q = 0; q < BATCHES_PER_BLOCK; ++q) acc += src[q];
        partial[(long long)blockIdx.x * 3 + tid] = acc;
    }
}

__global__ __launch_bounds__(256) void cayley_reduce_kernel(
    const float* __restrict__ partial, float* __restrict__ out,
    int nBlocks, int B) {
    __shared__ float rC[256], rE[256], rG[256];
    const int tid = threadIdx.x;
    float aC = 0.f, aE = 0.f, aG = 0.f;
    for (int j = tid; j < nBlocks; j += 256) {   // fixed partition -> deterministic
        aC += partial[j * 3 + 0];
        aE += partial[j * 3 + 1];
        aG += partial[j * 3 + 2];
    }
    rC[tid] = aC; rE[tid] = aE; rG[tid] = aG;
    __syncthreads();
    for (int s = 128; s > 0; s >>= 1) {
        if (tid < s) {
            rC[tid] += rC[tid + s];
            rE[tid] += rE[tid + s];
            rG[tid] += rG[tid + s];
        }
        __syncthreads();
    }
    if (tid == 0) {
        float invB = 1.0f / (float)B;
        out[0] = rC[0] * invB + 0.5f * rE[0] * invB + 0.1f * rG[0] * invB;
    }
}

extern "C" void kernel_launch(void* const* d_in, const int* in_sizes, int n_in,
                              void* d_out, int out_size, void* d_ws, size_t ws_size,
                              hipStream_t stream) {
    const float* X   = (const float*)d_in[0];
    float* out       = (float*)d_out;
    float* partial   = (float*)d_ws;

    const long long totalElems = (long long)in_sizes[0];   // B * 5 * 256
    const int B = (int)(totalElems / 1280);
    const long long totalRows = (long long)B * 5;
    const int nBlocks = (B + BATCHES_PER_BLOCK - 1) / BATCHES_PER_BLOCK;

    cayley_main_kernel<<<nBlocks, 256, 0, stream>>>(X, partial, B, totalRows);
    cayley_reduce_kernel<<<1, 256, 0, stream>>>(partial, out, nBlocks, B);
}